// NormXCorr_67138928771472
// MI455X (gfx1250) — compile-verified
//
#include <hip/hip_runtime.h>

typedef float v2f __attribute__((ext_vector_type(2)));
typedef float v4f __attribute__((ext_vector_type(4)));
typedef float v8f __attribute__((ext_vector_type(8)));

#define PS     5
#define PD     2
#define NP     25
#define EPSC   0.01f
#define Hn     64
#define Wn     24
#define Cn     32
#define Bn     16
#define WPAD   28
#define KSTEPS 7

// LDS layout (float offsets), 16B-aligned sections
#define XOFF   0                  // xwin: 6 rows x 28 (row 5 = zeros)
#define YOFF   (XOFF + 6*WPAD)    // ywin: 9 rows x 28
#define NUMOFF 420                // numoff[24] = sEc*Em   (420*4 % 16 == 0)
#define IESOFF 444                // invEsn[24] = 1/(25*Es)
#define FSOFF  468                // invFs[5*24]
#define EMOFF  588                // Em[24] (patch means of X)
#define SMEMSZ 612
#define ZSLOT  (XOFF + 5*WPAD)    // guaranteed-zero float (xwin row 5)

__device__ __forceinline__ constexpr int kmap(int k) {  // patch k -> row*WPAD + col
    return (k / PS) * WPAD + (k % PS);
}

__global__ __launch_bounds__(32)
void ncc_wmma_kernel(const float* __restrict__ X, const float* __restrict__ Y,
                     float* __restrict__ out)
{
    __shared__ __align__(16) float smem[SMEMSZ];

    const int lane = threadIdx.x;            // wave32
    const int bid  = blockIdx.x;
    const int h    = bid % Hn;
    const int c    = (bid / Hn) % Cn;
    const int b    = bid / (Hn * Cn);

    const float* Xbc = X + (size_t)(b * Cn + c) * (Hn * Wn);
    const float* Ybc = Y + (size_t)(b * Cn + c) * (Hn * Wn);

    // ---- staging: one padded column per lane (lanes 0..27) ----
    if (lane < WPAD) {
        const int col = lane - PD;
        const bool cok = (col >= 0) && (col < Wn);
#pragma unroll
        for (int r = 0; r < 6; ++r) {        // xwin rows; row 5 stays zero
            int row = h + r - PD;
            float v = 0.0f;
            if (r < 5 && cok && row >= 0 && row < Hn) v = Xbc[row * Wn + col];
            smem[XOFF + r * WPAD + lane] = v;
        }
#pragma unroll
        for (int r = 0; r < 9; ++r) {        // ywin rows: Y rows h-4 .. h+4
            int row = h + r - 2 * PD;
            float v = 0.0f;
            if (cok && row >= 0 && row < Hn) v = Ybc[row * Wn + col];
            smem[YOFF + r * WPAD + lane] = v;
        }
    }
    __syncthreads();

    // ---- X patch stats: lanes 0..23 handle x = lane ----
    if (lane < 24) {
        float s = 0.f;
#pragma unroll
        for (int kr = 0; kr < PS; ++kr)
#pragma unroll
            for (int kc = 0; kc < PS; ++kc)
                s += smem[XOFF + kr * WPAD + lane + kc];
        float mean = s * (1.0f / 25.0f);
        float ss = 0.f, sc = 0.f;
#pragma unroll
        for (int kr = 0; kr < PS; ++kr)
#pragma unroll
            for (int kc = 0; kc < PS; ++kc) {
                float d = smem[XOFF + kr * WPAD + lane + kc] - mean;
                ss += d * d;
                sc += d;
            }
        float es = sqrtf(ss * (1.0f / 24.0f)) + EPSC;  // ddof=1 sample std + eps
        smem[NUMOFF + lane] = sc * mean;               // sEc*Em (subtract from dot)
        smem[IESOFF + lane] = 1.0f / (25.0f * es);
        smem[EMOFF  + lane] = mean;
    }
    // ---- Y patch stats: 120 items (yo, i) ----
    for (int t = lane; t < 5 * 24; t += 32) {
        int yo = t / 24, i = t % 24;
        float s = 0.f;
#pragma unroll
        for (int kr = 0; kr < PS; ++kr)
#pragma unroll
            for (int kc = 0; kc < PS; ++kc)
                s += smem[YOFF + (yo + kr) * WPAD + i + kc];
        float mean = s * (1.0f / 25.0f);
        float ss = 0.f;
#pragma unroll
        for (int kr = 0; kr < PS; ++kr)
#pragma unroll
            for (int kc = 0; kc < PS; ++kc) {
                float d = smem[YOFF + (yo + kr) * WPAD + i + kc] - mean;
                ss += d * d;
            }
        float fs = sqrtf(ss * (1.0f / 24.0f)) + EPSC;
        smem[FSOFF + t] = 1.0f / fs;
    }
    __syncthreads();

    const int lh   = lane >> 4;     // lane half: selects K pair within a K=4 step
    const int lmod = lane & 15;

    // ---- A fragments (Ecen): lane holds row m = mt*16 + lmod, k = 4*ks + 2*lh + j.
    // k >= 25 reads xwin row 5 (zeros) -> value -Em; harmless since B is exactly 0 there.
    v2f afrag[2][KSTEPS];
#pragma unroll
    for (int mt = 0; mt < 2; ++mt) {
        int m  = mt * 16 + lmod;
        int mm = m < 24 ? m : 23;                // rows >= 24 are dead output lanes
        float mean = smem[EMOFF + mm];
#pragma unroll
        for (int ks = 0; ks < KSTEPS; ++ks) {
#pragma unroll
            for (int j = 0; j < 2; ++j) {
                const int klo = ks * 4 + j;      // lane half 0
                const int khi = klo + 2;         // lane half 1
                const int alo = XOFF + kmap(klo);   // klo<=25 -> row<=5, in-bounds
                const int ahi = XOFF + kmap(khi);   // khi<=27 -> row 5 zeros
                afrag[mt][ks][j] = smem[(lh ? ahi : alo) + mm] - mean;
            }
        }
    }

    // per-m-tile scale vectors (x = xb .. xb+7), hoisted out of yo loop.
    // mt==1,lh==1 reads past the 24-entry arrays (still inside smem); those lanes never store.
    v4f nof[2][2], ies[2][2];
#pragma unroll
    for (int mt = 0; mt < 2; ++mt) {
        int xb = mt * 16 + 8 * lh;
        nof[mt][0] = *(const v4f*)&smem[NUMOFF + xb];
        nof[mt][1] = *(const v4f*)&smem[NUMOFF + xb + 4];
        ies[mt][0] = *(const v4f*)&smem[IESOFF + xb];
        ies[mt][1] = *(const v4f*)&smem[IESOFF + xb + 4];
    }

    // ---- main loop: 5 vertical offsets x 2x2 output tiles x 7 K-steps ----
#pragma unroll
    for (int yo = 0; yo < PS; ++yo) {
#pragma unroll
        for (int nt = 0; nt < 2; ++nt) {
            const int n  = nt * 16 + lmod;
            const int nn = n < 24 ? n : 23;
            const int yb = YOFF + yo * WPAD + nn;
            // B fragment: raw Y patch, lane holds col n; k>=25 -> ZSLOT (exact zero)
            v2f bfrag[KSTEPS];
#pragma unroll
            for (int ks = 0; ks < KSTEPS; ++ks) {
#pragma unroll
                for (int j = 0; j < 2; ++j) {
                    const int klo = ks * 4 + j;
                    const int khi = klo + 2;
                    int alo = (klo < NP) ? (yb + kmap(klo)) : ZSLOT;
                    int ahi = (khi < NP) ? (yb + kmap(khi)) : ZSLOT;
                    bfrag[ks][j] = smem[lh ? ahi : alo];
                }
            }
            const float invfs = smem[FSOFF + yo * 24 + nn];
            const int i0 = nt * 16 + lmod;
#pragma unroll
            for (int mt = 0; mt < 2; ++mt) {
                v8f acc = {};
#pragma unroll
                for (int ks = 0; ks < KSTEPS; ++ks) {
                    acc = __builtin_amdgcn_wmma_f32_16x16x4_f32(
                        false, afrag[mt][ks], false, bfrag[ks],
                        (short)0, acc, false, false);
                }
                // D layout: VGPR v -> x = mt*16 + 8*lh + v, n = lmod (+nt*16).
                // Tile validity factors per tile -> one branch, two b128 stores.
                bool ok = true;
                if (mt == 1) ok = (lh == 0);
                if (nt == 1) ok = ok && (lmod < 8);
                if (ok) {
                    const int xb = mt * 16 + 8 * lh;
                    size_t obase = ((((size_t)b * (Cn * PS * Wn)
                                   + (size_t)(c * (PS * Wn) + yo * Wn + i0)) * Hn) + h) * Wn + xb;
                    v4f lo, hi;
#pragma unroll
                    for (int q = 0; q < 4; ++q) {
                        lo[q] = (acc[q]     - nof[mt][0][q]) * ies[mt][0][q] * invfs;
                        hi[q] = (acc[q + 4] - nof[mt][1][q]) * ies[mt][1][q] * invfs;
                    }
                    *(v4f*)(out + obase)     = lo;
                    *(v4f*)(out + obase + 4) = hi;
                }
            }
        }
    }
}

extern "C" void kernel_launch(void* const* d_in, const int* in_sizes, int n_in,
                              void* d_out, int out_size, void* d_ws, size_t ws_size,
                              hipStream_t stream) {
    const float* X = (const float*)d_in[0];
    const float* Y = (const float*)d_in[1];
    float* out = (float*)d_out;
    (void)in_sizes; (void)n_in; (void)out_size; (void)d_ws; (void)ws_size;
    dim3 grid(Bn * Cn * Hn);   // 32768 single-wave workgroups, one per (b, c, h)
    dim3 block(32);
    ncc_wmma_kernel<<<grid, block, 0, stream>>>(X, Y, out);
}